// SlideAttention_64484638982144
// MI455X (gfx1250) — compile-verified
//
#include <hip/hip_runtime.h>
#include <hip/hip_bf16.h>

// ---------------------------------------------------------------------------
// SlideAttention on MI455X (gfx1250), fp32 end-to-end.
//   X[b][256][64][64] -> qkv GEMM (WMMA f32 16x16x4, 3-tile register blocking)
//   -> slide attention (LDS haloed tiles) -> proj GEMM (WMMA, 4-tile blocking)
// Workspace: qkv (48 MiB) + attn out (16 MiB); both stay L2-resident (192 MB).
// HBM floor: read X 64 MiB + write out 64 MiB ~= 5.5 us @ 23.3 TB/s.
// ---------------------------------------------------------------------------

typedef float v2f __attribute__((ext_vector_type(2)));
typedef float v8f __attribute__((ext_vector_type(8)));

#define NB   16      // batch
#define CDIM 256     // input / output channels
#define HW   4096    // 64*64 pixels
#define WD   64
#define R3   192     // 3*CR qkv channels
#define NH   8
#define CPH  8
#define CR   64

__device__ __forceinline__ v8f wmma_f32_16x16x4(v2f a, v2f b, v8f c) {
  // 8 args: (neg_a, A, neg_b, B, c_mod, C, reuse_a, reuse_b)
  return __builtin_amdgcn_wmma_f32_16x16x4_f32(false, a, false, b, (short)0, c,
                                               false, false);
}

// ---------------------------------------------------------------------------
// Kernel A: qkv[b][r][p] = sum_c X[b][c][p] * Wq[r][c] + bq[r]
// One wave computes a 48(r) x 16(p) strip: 3 accumulator tiles share each
// B fragment (pixels of X), so X is read 4x instead of 12x through L2.
// ---------------------------------------------------------------------------
__global__ __launch_bounds__(256) void qkv_gemm_kernel(
    const float* __restrict__ X, const float* __restrict__ Wq,
    const float* __restrict__ bq, float* __restrict__ qkv) {
  const int wave = blockIdx.x * 8 + (threadIdx.x >> 5);
  const int lane = threadIdx.x & 31;
  const int hi   = lane >> 4;       // half-wave (selects K pair / M+8)
  const int ln   = lane & 15;
  const int rg   = wave & 3;        // 4 groups of 3 r-tiles (192 = 4*48)
  int t = wave >> 2;
  const int ptile = t & 255;        // 4096/16 pixel tiles
  const int b     = t >> 8;

  const int p = ptile * 16 + ln;    // B-matrix col (pixel)
  const float* xb = X + (size_t)b * CDIM * HW;
  const int row0 = rg * 48 + ln;    // A rows for the 3 tiles: row0 + 16*i

  v8f acc0 = {0.f,0.f,0.f,0.f,0.f,0.f,0.f,0.f};
  v8f acc1 = acc0, acc2 = acc0;
#pragma unroll 4
  for (int k0 = 0; k0 < CDIM; k0 += 4) {
    const int ka = k0 + 2 * hi;     // lanes 0-15: K=k0,k0+1; 16-31: +2
    v2f bb; bb.x = xb[ka * HW + p]; bb.y = xb[(ka + 1) * HW + p];
    v2f a0; a0.x = Wq[(row0     ) * CDIM + ka]; a0.y = Wq[(row0     ) * CDIM + ka + 1];
    v2f a1; a1.x = Wq[(row0 + 16) * CDIM + ka]; a1.y = Wq[(row0 + 16) * CDIM + ka + 1];
    v2f a2; a2.x = Wq[(row0 + 32) * CDIM + ka]; a2.y = Wq[(row0 + 32) * CDIM + ka + 1];
    acc0 = wmma_f32_16x16x4(a0, bb, acc0);
    acc1 = wmma_f32_16x16x4(a1, bb, acc1);
    acc2 = wmma_f32_16x16x4(a2, bb, acc2);
  }
  float* outb = qkv + (size_t)b * R3 * HW + ptile * 16 + ln;
#pragma unroll
  for (int g = 0; g < 8; ++g) {
    const int r = rg * 48 + g + 8 * hi;    // C row M = g (+8 upper half)
    outb[(r     ) * HW] = acc0[g] + bq[r     ];
    outb[(r + 16) * HW] = acc1[g] + bq[r + 16];
    outb[(r + 32) * HW] = acc2[g] + bq[r + 32];
  }
}

// ---------------------------------------------------------------------------
// Kernel B: slide attention.  One block = one (batch, 16x16 tile, head).
// k_{c,j} = kk[p+off_j] + dwconv1(kk)_{c,j} + b0[c*9+j] + b1[c*9+j] (+rpb via qsum)
// ---------------------------------------------------------------------------
__global__ __launch_bounds__(256) void slide_attn_kernel(
    const float* __restrict__ qkv, const float* __restrict__ b0,
    const float* __restrict__ w1, const float* __restrict__ b1,
    const float* __restrict__ rpb, float* __restrict__ out64) {
  __shared__ float kk_s[CPH][18][18];
  __shared__ float vv_s[CPH][18][18];
  __shared__ float w1_s[72 * 9];
  __shared__ float bsum_s[72];
  __shared__ float rpb_s[72];

  const int tid = threadIdx.x;
  const int b   = blockIdx.x / 128;         // 16 tiles * 8 heads per batch
  const int rem = blockIdx.x % 128;
  const int tile = rem >> 3;
  const int n    = rem & 7;                 // head
  const int h0 = (tile >> 2) * 16;
  const int w0 = (tile & 3) * 16;
  const int lh = tid >> 4;
  const int lw = tid & 15;
  const int p  = (h0 + lh) * WD + (w0 + lw);

  for (int i = tid; i < 648; i += 256) w1_s[i] = w1[i];
  for (int i = tid; i < 72; i += 256) {
    bsum_s[i] = b0[i] + b1[i];
    rpb_s[i]  = rpb[i];
  }

  const float* qb = qkv + (size_t)b * R3 * HW;
  // Load haloed kk / vv tiles (18x18 x 8 channels) for this head.
  for (int i = tid; i < CPH * 324; i += 256) {
    const int c = i / 324;
    const int s = i - c * 324;
    const int ih = s / 18, iw = s - ih * 18;
    const int gh = h0 - 1 + ih, gw = w0 - 1 + iw;
    float kv = 0.f, vv = 0.f;
    if (gh >= 0 && gh < WD && gw >= 0 && gw < WD) {
      const int gp = gh * WD + gw;
      kv = qb[(n * 24 + 8 + c) * HW + gp];
      vv = qb[(n * 24 + 16 + c) * HW + gp];
    }
    kk_s[c][ih][iw] = kv;
    vv_s[c][ih][iw] = vv;
  }
  __syncthreads();

  const float scale = 0.17677669529663687f;   // (256/8)^-0.5
  float logits[9];
#pragma unroll
  for (int j = 0; j < 9; ++j) logits[j] = 0.f;
  float qsum = 0.f;

#pragma unroll
  for (int c = 0; c < CPH; ++c) {
    const float q = qb[(n * 24 + c) * HW + p] * scale;
    qsum += q;
    float nb[9];
#pragma unroll
    for (int t = 0; t < 9; ++t) nb[t] = kk_s[c][lh + t / 3][lw + t % 3];
#pragma unroll
    for (int j = 0; j < 9; ++j) {
      const int o = c * 9 + j;
      float kj = nb[j] + bsum_s[o];
#pragma unroll
      for (int t = 0; t < 9; ++t) kj = fmaf(w1_s[o * 9 + t], nb[t], kj);
      logits[j] = fmaf(q, kj, logits[j]);
    }
  }
  // rpb is per (head, j), constant over c:  q . rpb_j = rpb_j * sum_c q_c
#pragma unroll
  for (int j = 0; j < 9; ++j) logits[j] = fmaf(qsum, rpb_s[n * 9 + j], logits[j]);

  float mx = logits[0];
#pragma unroll
  for (int j = 1; j < 9; ++j) mx = fmaxf(mx, logits[j]);
  float aw[9], ssum = 0.f;
#pragma unroll
  for (int j = 0; j < 9; ++j) { aw[j] = __expf(logits[j] - mx); ssum += aw[j]; }
  const float inv = 1.f / ssum;
#pragma unroll
  for (int j = 0; j < 9; ++j) aw[j] *= inv;

  float* ob = out64 + (size_t)(b * CR + n * CPH) * HW + p;
#pragma unroll
  for (int c = 0; c < CPH; ++c) {
    float nb[9];
#pragma unroll
    for (int t = 0; t < 9; ++t) nb[t] = vv_s[c][lh + t / 3][lw + t % 3];
    float oc = 0.f;
#pragma unroll
    for (int j = 0; j < 9; ++j) {
      const int o = c * 9 + j;
      float vj = nb[j] + bsum_s[o];
#pragma unroll
      for (int t = 0; t < 9; ++t) vj = fmaf(w1_s[o * 9 + t], nb[t], vj);
      oc = fmaf(aw[j], vj, oc);
    }
    ob[c * HW] = oc;
  }
}

// ---------------------------------------------------------------------------
// Kernel C: out[b][co][p] = sum_cr in64[b][cr][p] * Wp[co][cr] + bp[co], K=64
// One wave computes a 64(co) x 16(p) strip: 4 accumulator tiles share each
// B fragment, so in64 is read 4x instead of 16x through L2.
// ---------------------------------------------------------------------------
__global__ __launch_bounds__(256) void proj_gemm_kernel(
    const float* __restrict__ in64, const float* __restrict__ Wp,
    const float* __restrict__ bp, float* __restrict__ out) {
  const int wave = blockIdx.x * 8 + (threadIdx.x >> 5);
  const int lane = threadIdx.x & 31;
  const int hi   = lane >> 4;
  const int ln   = lane & 15;
  const int cg   = wave & 3;            // 4 groups of 4 co-tiles (256 = 4*64)
  int t = wave >> 2;
  const int ptile = t & 255;
  const int b     = t >> 8;

  const int p = ptile * 16 + ln;
  const float* ib = in64 + (size_t)b * CR * HW;
  const int row0 = cg * 64 + ln;        // A rows: row0 + 16*i

  v8f acc0 = {0.f,0.f,0.f,0.f,0.f,0.f,0.f,0.f};
  v8f acc1 = acc0, acc2 = acc0, acc3 = acc0;
#pragma unroll 4
  for (int k0 = 0; k0 < CR; k0 += 4) {
    const int ka = k0 + 2 * hi;
    v2f bb; bb.x = ib[ka * HW + p]; bb.y = ib[(ka + 1) * HW + p];
    v2f a0; a0.x = Wp[(row0     ) * CR + ka]; a0.y = Wp[(row0     ) * CR + ka + 1];
    v2f a1; a1.x = Wp[(row0 + 16) * CR + ka]; a1.y = Wp[(row0 + 16) * CR + ka + 1];
    v2f a2; a2.x = Wp[(row0 + 32) * CR + ka]; a2.y = Wp[(row0 + 32) * CR + ka + 1];
    v2f a3; a3.x = Wp[(row0 + 48) * CR + ka]; a3.y = Wp[(row0 + 48) * CR + ka + 1];
    acc0 = wmma_f32_16x16x4(a0, bb, acc0);
    acc1 = wmma_f32_16x16x4(a1, bb, acc1);
    acc2 = wmma_f32_16x16x4(a2, bb, acc2);
    acc3 = wmma_f32_16x16x4(a3, bb, acc3);
  }
  float* ob = out + (size_t)b * CDIM * HW + ptile * 16 + ln;
#pragma unroll
  for (int g = 0; g < 8; ++g) {
    const int r = cg * 64 + g + 8 * hi;
    ob[(r     ) * HW] = acc0[g] + bp[r     ];
    ob[(r + 16) * HW] = acc1[g] + bp[r + 16];
    ob[(r + 32) * HW] = acc2[g] + bp[r + 32];
    ob[(r + 48) * HW] = acc3[g] + bp[r + 48];
  }
}

// ---------------------------------------------------------------------------
extern "C" void kernel_launch(void* const* d_in, const int* in_sizes, int n_in,
                              void* d_out, int out_size, void* d_ws, size_t ws_size,
                              hipStream_t stream) {
  const float* x           = (const float*)d_in[0];
  // d_in[1..4]: H22, W22, relative_pos_index, relative_coords_table (unused)
  const float* qkv_w       = (const float*)d_in[5];
  const float* qkv_b       = (const float*)d_in[6];
  const float* dep_conv_b  = (const float*)d_in[7];
  const float* dep_conv1_w = (const float*)d_in[8];
  const float* dep_conv1_b = (const float*)d_in[9];
  const float* rpb_table   = (const float*)d_in[10];
  const float* proj_w      = (const float*)d_in[11];
  const float* proj_b      = (const float*)d_in[12];

  float* qkv   = (float*)d_ws;                       // 16*192*4096 f32 = 48 MiB
  float* out64 = qkv + (size_t)NB * R3 * HW;         // 16*64*4096  f32 = 16 MiB

  // A: 16 batches * 256 pixel-tiles * 4 r-groups = 16384 waves / 8 per block
  qkv_gemm_kernel<<<2048, 256, 0, stream>>>(x, qkv_w, qkv_b, qkv);
  // B: 16 batches * 16 tiles * 8 heads = 2048 blocks
  slide_attn_kernel<<<2048, 256, 0, stream>>>(qkv, dep_conv_b, dep_conv1_w,
                                              dep_conv1_b, rpb_table, out64);
  // C: 16 batches * 256 pixel-tiles * 4 co-groups = 16384 waves / 8 per block
  proj_gemm_kernel<<<2048, 256, 0, stream>>>(out64, proj_w, proj_b, (float*)d_out);
}